// SNNLocalConvCell_74526272520629
// MI455X (gfx1250) — compile-verified
//
#include <hip/hip_runtime.h>

// Problem constants (match reference)
#define B_    128
#define CI_   16
#define CO_   32
#define H_    32
#define W_    32
#define K_    3
#define HO_   30
#define WO_   30
#define P_    (HO_*WO_)          // 900
#define KC_   (CI_*K_*K_)        // 144
#define NELEM (B_*CO_*P_)        // 3,686,400
#define NSPLIT 16
#define PQ_   (P_/4)             // 225 float4 per [b,o] row

typedef __attribute__((ext_vector_type(2))) float v2f;
typedef __attribute__((ext_vector_type(8))) float v8f;

// ---------------------------------------------------------------------------
// Kernel 1: per-location GEMMs (conv K=144, rec K=32) via v_wmma_f32_16x16x4_f32
// grid = P_ blocks, 256 threads = 8 waves; wave w owns batches [16w,16w+16)
// ---------------------------------------------------------------------------
__global__ __launch_bounds__(256) void k_gemm(
    const float* __restrict__ ff,      // [B,CI,H,W]
    const float* __restrict__ spk,     // [B,CO,HO,WO]
    const float* __restrict__ weight,  // [1,CO,CI,HO,WO,9]
    const float* __restrict__ cbias,   // [1,CO,HO,WO]
    const float* __restrict__ lrec,    // [P,CO,CO]
    float* __restrict__ conv_ws,       // [B,CO,P]
    float* __restrict__ rec_ws)        // [B,CO,P]
{
    // A-fragment staging: [kpair][m][2] (float2 rows of the K dimension)
    __shared__ alignas(16) float Wf[(KC_/2) * (CO_*2 + 2)];   // 72*66 floats
    __shared__ alignas(16) float Lf[(CO_/2) * (CO_*2 + 2)];   // 16*66 floats
    __shared__ int   offI[KC_];                                // patch offsets
    __shared__ float Cb[CO_];                                  // conv bias at p

    const int p   = blockIdx.x;
    const int h0  = p / WO_;
    const int w0  = p % WO_;
    const int tid = threadIdx.x;

    // Stage conv weights transposed to [kf][o] (float2-interleaved over kf)
    // via async global->LDS scatter (ASYNCcnt).  4608 = 18*256: no divergence,
    // EXEC is all-ones at every issue as the async op requires.
    for (int idx = tid; idx < CO_*KC_; idx += 256) {
        int o  = idx / KC_;
        int kf = idx % KC_;
        int c  = kf / 9, r = kf % 9;
        const float* gsrc = &weight[(o*CI_ + c)*(P_*9) + p*9 + r];
        unsigned ldsoff = (unsigned)(unsigned long long)(void*)
                          &Wf[(kf >> 1)*66 + o*2 + (kf & 1)];
        asm volatile("global_load_async_to_lds_b32 %0, %1, off"
                     :: "v"(ldsoff), "v"(gsrc) : "memory");
    }
    // Stage local_rec[p] transposed to [j][i], same async path (1024 = 4*256)
    for (int idx = tid; idx < CO_*CO_; idx += 256) {
        int i = idx / CO_;
        int j = idx % CO_;
        const float* gsrc = &lrec[p*CO_*CO_ + i*CO_ + j];
        unsigned ldsoff = (unsigned)(unsigned long long)(void*)
                          &Lf[(j >> 1)*66 + i*2 + (j & 1)];
        asm volatile("global_load_async_to_lds_b32 %0, %1, off"
                     :: "v"(ldsoff), "v"(gsrc) : "memory");
    }
    // Patch offset table + per-location bias (regular path, overlaps async)
    for (int kf = tid; kf < KC_; kf += 256) {
        int c = kf / 9, r = kf % 9;
        offI[kf] = (c*H_ + h0 + r/3)*W_ + (w0 + r%3);
    }
    if (tid < CO_) Cb[tid] = cbias[tid*P_ + p];
    asm volatile("s_wait_asynccnt 0" ::: "memory");
    __syncthreads();

    const int wv   = tid >> 5;        // wave id 0..7
    const int lane = tid & 31;
    const int half = lane >> 4;       // K-pair selector per WMMA A/B layout
    const int lm   = lane & 15;
    const int b    = wv*16 + lm;      // this lane's batch column (N index)

    const v2f* W2 = (const v2f*)Wf;   // [72][33] float2, row stride 33
    const v2f* L2 = (const v2f*)Lf;   // [16][33] float2

    const int bb_ff  = b * (CI_*H_*W_);
    const int bb_spk = b * (CO_*P_);

    // ---- conv GEMM: M=32 (2 tiles), N=16, K=144 (36 steps of 4) ----
    v8f c0 = {}; v8f c1 = {};
    for (int ks = 0; ks < KC_/4; ++ks) {
        int kp  = 2*ks + half;                 // float2 row of K
        v2f a0  = W2[kp*33 + lm];              // channels 0..15
        v2f a1  = W2[kp*33 + 16 + lm];         // channels 16..31
        int kf0 = kp*2;
        v2f bf;
        bf.x = ff[bb_ff + offI[kf0]];
        bf.y = ff[bb_ff + offI[kf0 + 1]];
        c0 = __builtin_amdgcn_wmma_f32_16x16x4_f32(false, a0, false, bf,
                                                   (short)0, c0, false, false);
        c1 = __builtin_amdgcn_wmma_f32_16x16x4_f32(false, a1, false, bf,
                                                   (short)0, c1, false, false);
    }

    // ---- rec GEMM: M=32, N=16, K=32 (8 steps of 4) ----
    v8f r0 = {}; v8f r1 = {};
    for (int ks = 0; ks < CO_/4; ++ks) {
        int kp = 2*ks + half;
        v2f a0 = L2[kp*33 + lm];
        v2f a1 = L2[kp*33 + 16 + lm];
        int j0 = kp*2;
        v2f bf;
        bf.x = spk[bb_spk + j0*P_ + p];
        bf.y = spk[bb_spk + (j0 + 1)*P_ + p];
        r0 = __builtin_amdgcn_wmma_f32_16x16x4_f32(false, a0, false, bf,
                                                   (short)0, r0, false, false);
        r1 = __builtin_amdgcn_wmma_f32_16x16x4_f32(false, a1, false, bf,
                                                   (short)0, r1, false, false);
    }

    // ---- store: C/D layout: VGPR e holds M = e + 8*half, N = lane%16 ----
    for (int e = 0; e < 8; ++e) {
        int o0 = e + 8*half;
        int o1 = o0 + 16;
        int base0 = (b*CO_ + o0)*P_ + p;
        int base1 = (b*CO_ + o1)*P_ + p;
        conv_ws[base0] = c0[e] + Cb[o0];
        rec_ws [base0] = r0[e];
        conv_ws[base1] = c1[e] + Cb[o1];
        rec_ws [base1] = r1[e];
    }
}

// ---------------------------------------------------------------------------
// Kernel 2a: per-channel partial BN sums, split over batch for parallelism
// grid = (CO_, NSPLIT); float4 reads (each [b,o] row = 225 aligned float4s)
// ---------------------------------------------------------------------------
__global__ __launch_bounds__(256) void k_stats1(
    const float4* __restrict__ conv4,
    float* __restrict__ partials)      // [CO_][NSPLIT][2]
{
    __shared__ float s1[256];
    __shared__ float s2[256];
    const int o   = blockIdx.x;
    const int s   = blockIdx.y;
    const int tid = threadIdx.x;
    const int BSP = B_ / NSPLIT;       // 8 batches per split
    float sum = 0.f, sq = 0.f;
    for (int idx = tid; idx < BSP*PQ_; idx += 256) {
        int bb = s*BSP + idx / PQ_;
        int pq = idx % PQ_;
        float4 v = conv4[(bb*CO_ + o)*PQ_ + pq];
        sum += v.x + v.y + v.z + v.w;
        sq  += v.x*v.x + v.y*v.y + v.z*v.z + v.w*v.w;
    }
    s1[tid] = sum; s2[tid] = sq;
    __syncthreads();
    for (int st = 128; st > 0; st >>= 1) {
        if (tid < st) { s1[tid] += s1[tid+st]; s2[tid] += s2[tid+st]; }
        __syncthreads();
    }
    if (tid == 0) {
        partials[(o*NSPLIT + s)*2 + 0] = s1[0];
        partials[(o*NSPLIT + s)*2 + 1] = s2[0];
    }
}

// ---------------------------------------------------------------------------
// Kernel 2b: finalize BN scale/shift per channel
// ---------------------------------------------------------------------------
__global__ void k_stats2(
    const float* __restrict__ partials,
    const float* __restrict__ gamma,
    const float* __restrict__ beta,
    float* __restrict__ ss)            // [0..CO): scale, [CO..2CO): shift
{
    int o = threadIdx.x;
    if (o < CO_) {
        float s = 0.f, q = 0.f;
        for (int i = 0; i < NSPLIT; ++i) {
            s += partials[(o*NSPLIT + i)*2 + 0];
            q += partials[(o*NSPLIT + i)*2 + 1];
        }
        const float inv_n = 1.0f / (float)(B_*P_);
        float mean = s * inv_n;
        float var  = q * inv_n - mean*mean;   // biased variance
        float sc   = gamma[o] * rsqrtf(var + 1e-5f);
        ss[o]       = sc;
        ss[CO_ + o] = beta[o] - mean*sc;
    }
}

// ---------------------------------------------------------------------------
// Kernel 3: fused BN-apply + rec-add + adaptive LIF update, float4 vectorized
// (NELEM % 4 == 0; all tensors contiguous and 16B aligned)
// ---------------------------------------------------------------------------
__global__ __launch_bounds__(256) void k_lif(
    const float4* __restrict__ fb,
    const float4* __restrict__ soma,
    const float4* __restrict__ spk,
    const float4* __restrict__ acur,
    const float4* __restrict__ bt,
    const float4* __restrict__ conv4,
    const float4* __restrict__ rec4,
    const float* __restrict__ ss,
    const float* __restrict__ tau_m,
    const float* __restrict__ tau_adp,
    const float* __restrict__ tau_a,
    float4* __restrict__ out)
{
    int v = blockIdx.x * 256 + threadIdx.x;
    if (v >= NELEM/4) return;
    const int idx = v * 4;

    float4 cv = conv4[v], rc = rec4[v];
    float4 f  = fb[v],   sm = soma[v], sp4 = spk[v];
    float4 ac = acur[v], b4 = bt[v];

    float4 o_soma, o_spike, o_an, o_bn;
    float* pcv = &cv.x;  float* prc = &rc.x;  float* pf  = &f.x;
    float* psm = &sm.x;  float* psp = &sp4.x; float* pac = &ac.x;
    float* pb  = &b4.x;
    float* r0 = &o_soma.x; float* r1 = &o_spike.x;
    float* r2 = &o_an.x;   float* r3 = &o_bn.x;

    #pragma unroll
    for (int j = 0; j < 4; ++j) {
        int t  = idx + j;
        int pp = t % P_;
        int o  = (t / P_) % CO_;
        int op = o*P_ + pp;

        float conv_x = pcv[j]*ss[o] + ss[CO_ + o] + prc[j];
        float alpha = expf(-0.5f / tau_m[op]);
        float rho   = expf(-0.5f / tau_adp[op]);
        float eta   = expf(-0.5f / tau_a[op]);

        float sp  = psp[j];
        float bn  = rho*pb[j] + (1.0f - rho)*sp;
        float thr = 0.1f + 1.8f*bn;
        float an  = eta*pac[j] + pf[j];
        float sig = 1.0f / (1.0f + expf(-an));
        float sn  = alpha*psm[j] + (sig - 0.5f) + conv_x - thr*sp;

        r0[j] = sn;
        r1[j] = (sn - thr > 0.0f) ? 1.0f : 0.0f;
        r2[j] = an;
        r3[j] = bn;
    }

    out[v]                 = o_soma;
    out[NELEM/4 + v]       = o_spike;
    out[2*(NELEM/4) + v]   = o_an;
    out[3*(NELEM/4) + v]   = o_bn;
}

// ---------------------------------------------------------------------------
extern "C" void kernel_launch(void* const* d_in, const int* in_sizes, int n_in,
                              void* d_out, int out_size, void* d_ws, size_t ws_size,
                              hipStream_t stream) {
    const float* ff      = (const float*)d_in[0];
    const float* fb      = (const float*)d_in[1];
    const float* soma    = (const float*)d_in[2];
    const float* spk     = (const float*)d_in[3];
    const float* acur    = (const float*)d_in[4];
    const float* bt      = (const float*)d_in[5];
    const float* weight  = (const float*)d_in[6];
    const float* cbias   = (const float*)d_in[7];
    const float* lrec    = (const float*)d_in[8];
    const float* gamma   = (const float*)d_in[9];
    const float* beta    = (const float*)d_in[10];
    const float* tau_m   = (const float*)d_in[11];
    const float* tau_adp = (const float*)d_in[12];
    const float* tau_a   = (const float*)d_in[13];

    float* out      = (float*)d_out;
    float* conv_ws  = (float*)d_ws;
    float* rec_ws   = conv_ws + NELEM;
    float* ss       = rec_ws + NELEM;          // 2*CO_ floats
    float* partials = ss + 2*CO_;              // CO_*NSPLIT*2 floats

    k_gemm<<<P_, 256, 0, stream>>>(ff, spk, weight, cbias, lrec, conv_ws, rec_ws);
    k_stats1<<<dim3(CO_, NSPLIT), 256, 0, stream>>>((const float4*)conv_ws, partials);
    k_stats2<<<1, 32, 0, stream>>>(partials, gamma, beta, ss);
    k_lif<<<(NELEM/4 + 255)/256, 256, 0, stream>>>(
        (const float4*)fb, (const float4*)soma, (const float4*)spk,
        (const float4*)acur, (const float4*)bt,
        (const float4*)conv_ws, (const float4*)rec_ws, ss,
        tau_m, tau_adp, tau_a, (float4*)out);
}